// GravityPooling_49675591745656
// MI455X (gfx1250) — compile-verified
//
#include <hip/hip_runtime.h>
#include <math.h>

// GravityPooling for MI455X (gfx1250, wave32).
// kNN phase: V_WMMA_F32_16X16X4_F32 with A=(cx,cy,cz,|c|^2), B=(-2qx,-2qy,-2qz,1)
// => each WMMA tile directly yields the selection key |c|^2 - 2 c.q.
// Candidates double-buffered in LDS via GLOBAL_LOAD_ASYNC_TO_LDS_B64 (ASYNCcnt),
// stored split as (x,y) / (z,|c|^2) so each lane's A operand is one ds_load_b64.

#define NB    4
#define NPT   8192
#define NKEEP 4096
#define KNN   5
#define CHUNK 1024          // candidates per LDS buffer
#define SLOTS (CHUNK / 256) // async copies per thread per chunk (x2 halves)

typedef float v2f __attribute__((ext_vector_type(2)));
typedef float v8f __attribute__((ext_vector_type(8)));

// ---------------- CDNA5 async global->LDS helpers ----------------

__device__ __forceinline__ unsigned int lds_addr(const void* p) {
    // generic shared pointer: low 32 bits are the LDS byte offset (ISA 10.2 aperture rules)
    return (unsigned int)(size_t)p;
}

__device__ __forceinline__ void async_ld_b64(unsigned int lds, const void* gaddr) {
    asm volatile("global_load_async_to_lds_b64 %0, %1, off"
                 :: "v"(lds), "v"(gaddr)
                 : "memory");
}

__device__ __forceinline__ void wait_async_8() {
    asm volatile("s_wait_asynccnt 0x8" ::: "memory");
}
__device__ __forceinline__ void wait_async_0() {
    asm volatile("s_wait_asynccnt 0x0" ::: "memory");
}

// ---------------- gravity ----------------

__global__ void gp_centroid_in(const float* __restrict__ c, float* __restrict__ cent) {
    __shared__ float sx[256], sy[256], sz[256];
    const int b = blockIdx.x;
    const int t = threadIdx.x;
    const float* cb = c + (size_t)b * NPT * 3;
    float ax = 0.f, ay = 0.f, az = 0.f;
    for (int i = t; i < NPT; i += 256) {
        ax += cb[i * 3 + 0]; ay += cb[i * 3 + 1]; az += cb[i * 3 + 2];
    }
    sx[t] = ax; sy[t] = ay; sz[t] = az;
    __syncthreads();
    for (int s = 128; s > 0; s >>= 1) {
        if (t < s) { sx[t] += sx[t + s]; sy[t] += sy[t + s]; sz[t] += sz[t + s]; }
        __syncthreads();
    }
    if (t == 0) {
        cent[b * 4 + 0] = sx[0] * (1.0f / NPT);
        cent[b * 4 + 1] = sy[0] * (1.0f / NPT);
        cent[b * 4 + 2] = sz[0] * (1.0f / NPT);
    }
}

__global__ void gp_centroid4(const float4* __restrict__ c, float* __restrict__ cent) {
    __shared__ float sx[256], sy[256], sz[256];
    const int b = blockIdx.x;
    const int t = threadIdx.x;
    const float4* cb = c + (size_t)b * NPT;
    float ax = 0.f, ay = 0.f, az = 0.f;
    for (int i = t; i < NPT; i += 256) {
        float4 p = cb[i];
        ax += p.x; ay += p.y; az += p.z;
    }
    sx[t] = ax; sy[t] = ay; sz[t] = az;
    __syncthreads();
    for (int s = 128; s > 0; s >>= 1) {
        if (t < s) { sx[t] += sx[t + s]; sy[t] += sy[t + s]; sz[t] += sz[t + s]; }
        __syncthreads();
    }
    if (t == 0) {
        cent[b * 4 + 0] = sx[0] * (1.0f / NPT);
        cent[b * 4 + 1] = sy[0] * (1.0f / NPT);
        cent[b * 4 + 2] = sz[0] * (1.0f / NPT);
    }
}

__device__ __forceinline__ float4 gravity_step(float px, float py, float pz,
                                               float cx, float cy, float cz) {
    float dx = cx - px, dy = cy - py, dz = cz - pz;
    float dist = sqrtf(dx * dx + dy * dy + dz * dz);
    // t=1, delta=1: p += 0.5*(dist-1)*dir/dist
    float f = 0.5f * (dist - 1.0f) / dist;
    float nx = px + f * dx, ny = py + f * dy, nz = pz + f * dz;
    float4 r; r.x = nx; r.y = ny; r.z = nz; r.w = nx * nx + ny * ny + nz * nz;
    return r;
}

__global__ void gp_update_in(const float* __restrict__ cin, const float* __restrict__ cent,
                             float4* __restrict__ cand4) {
    const int b = blockIdx.y;
    const int i = blockIdx.x * 256 + threadIdx.x;
    const float* p = cin + ((size_t)b * NPT + i) * 3;
    cand4[(size_t)b * NPT + i] =
        gravity_step(p[0], p[1], p[2], cent[b * 4 + 0], cent[b * 4 + 1], cent[b * 4 + 2]);
}

__global__ void gp_update4(float4* __restrict__ cand4, const float* __restrict__ cent) {
    const int b = blockIdx.y;
    const int i = blockIdx.x * 256 + threadIdx.x;
    float4 p = cand4[(size_t)b * NPT + i];
    cand4[(size_t)b * NPT + i] =
        gravity_step(p.x, p.y, p.z, cent[b * 4 + 0], cent[b * 4 + 1], cent[b * 4 + 2]);
}

// ---------------- kNN (WMMA) + density ----------------

__device__ __forceinline__ void top5_insert(float key, int idx,
    float& k0, float& k1, float& k2, float& k3, float& k4,
    int& i0, int& i1, int& i2, int& i3, int& i4) {
    bool c4 = key < k4;
    bool c3 = key < k3;
    bool c2 = key < k2;
    bool c1 = key < k1;
    bool c0 = key < k0;
    k4 = c4 ? (c3 ? k3 : key) : k4;  i4 = c4 ? (c3 ? i3 : idx) : i4;
    k3 = c3 ? (c2 ? k2 : key) : k3;  i3 = c3 ? (c2 ? i2 : idx) : i3;
    k2 = c2 ? (c1 ? k1 : key) : k2;  i2 = c2 ? (c1 ? i1 : idx) : i2;
    k1 = c1 ? (c0 ? k0 : key) : k1;  i1 = c1 ? (c0 ? i0 : idx) : i1;
    k0 = c0 ? key : k0;              i0 = c0 ? idx : i0;
}

// Block: 256 threads = 8 waves; each wave owns 16 query rows (WMMA N dim).
// Grid: (NPT/128, NB).
__global__ void gp_topk(const float4* __restrict__ cand4,
                        float* __restrict__ dens,
                        int* __restrict__ nbr) {
    __shared__ float2 scA[2][CHUNK];   // (x, y)        per candidate
    __shared__ float2 scB[2][CHUNK];   // (z, |c|^2)    per candidate
    __shared__ float  smk[8][32][KNN];
    __shared__ int    smi[8][32][KNN];

    const int b    = blockIdx.y;
    const int tid  = threadIdx.x;
    const int wave = tid >> 5;
    const int lane = tid & 31;
    const int half = lane >> 4;             // 0: C rows M=0..7  1: M=8..15
    const int lm   = lane & 15;
    const float4* cb4 = cand4 + (size_t)b * NPT;
    const int qbase = blockIdx.x * 128 + wave * 16;

    // B operand (4x16 f32): lane n<16 -> K=0,1 ; lane n+16 -> K=2,3
    float4 q4 = cb4[qbase + lm];
    v2f Bm;
    Bm.x = half ? (-2.0f * q4.z) : (-2.0f * q4.x);
    Bm.y = half ? 1.0f           : (-2.0f * q4.y);

    float k0 = 1e30f, k1 = 1e30f, k2 = 1e30f, k3 = 1e30f, k4 = 1e30f;
    int   i0 = 0, i1 = 0, i2 = 0, i3 = 0, i4 = 0;

    const int nchunks = NPT / CHUNK;

    // prologue: async-stage chunk 0 into buffer 0 (8 async ops per wave)
#pragma unroll
    for (int s = 0; s < SLOTS; ++s) {
        const int idx = s * 256 + tid;
        const char* g = (const char*)&cb4[idx];
        async_ld_b64(lds_addr(&scA[0][idx]), g);
        async_ld_b64(lds_addr(&scB[0][idx]), g + 8);
    }

    for (int c = 0; c < nchunks; ++c) {
        const int cur = c & 1;
        const int cb  = c * CHUNK;

        if (c + 1 < nchunks) {
            // stream next chunk while we compute on the current one
            const int nxt = cur ^ 1;
            const int nb2 = cb + CHUNK;
#pragma unroll
            for (int s = 0; s < SLOTS; ++s) {
                const int idx = s * 256 + tid;
                const char* g = (const char*)&cb4[nb2 + idx];
                async_ld_b64(lds_addr(&scA[nxt][idx]), g);
                async_ld_b64(lds_addr(&scB[nxt][idx]), g + 8);
            }
            wait_async_8();   // current chunk's 8 ops retired (in-order), next 8 in flight
        } else {
            wait_async_0();
        }
        __syncthreads();      // all waves' copies for this buffer visible

        // A operand (16x4 f32): lane m<16 needs (x,y)=K0,1 ; lane m+16 needs (z,n2)=K2,3
        const float2* curA = half ? &scB[cur][0] : &scA[cur][0];

        for (int t = 0; t < CHUNK / 16; ++t) {
            float2 a2 = curA[t * 16 + lm];
            v2f Am; Am.x = a2.x; Am.y = a2.y;
            v8f C = {};
            C = __builtin_amdgcn_wmma_f32_16x16x4_f32(
                    false, Am, false, Bm, (short)0, C, false, false);
            // lane holds key(q = qbase+lm, cand = jb + r), r = 0..7
            const int jb = cb + t * 16 + half * 8;
#pragma unroll
            for (int r = 0; r < 8; ++r) {
                top5_insert(C[r], jb + r, k0, k1, k2, k3, k4, i0, i1, i2, i3, i4);
            }
        }
        __syncthreads();      // buffer will be overwritten by the next issue
    }

    // merge the two half-wave top-5 lists for each query row
    smk[wave][lane][0] = k0; smk[wave][lane][1] = k1; smk[wave][lane][2] = k2;
    smk[wave][lane][3] = k3; smk[wave][lane][4] = k4;
    smi[wave][lane][0] = i0; smi[wave][lane][1] = i1; smi[wave][lane][2] = i2;
    smi[wave][lane][3] = i3; smi[wave][lane][4] = i4;
    __syncthreads();

    if (lane < 16) {
        int pa = 0, pb = 0;
        int sel[KNN];
#pragma unroll
        for (int s = 0; s < KNN; ++s) {
            float ka = smk[wave][lane][pa];
            float kb = smk[wave][lane + 16][pb];
            int   ia = smi[wave][lane][pa];
            int   ib = smi[wave][lane + 16][pb];
            bool takeA = (ka < kb) || (ka == kb && ia < ib);
            sel[s] = takeA ? ia : ib;
            pa += takeA ? 1 : 0;
            pb += takeA ? 0 : 1;
        }
        // density = sum over kxk pairwise distances = 2 * sum_{a<b} dist
        float4 p[KNN];
#pragma unroll
        for (int s = 0; s < KNN; ++s) p[s] = cb4[sel[s]];
        float sum = 0.f;
#pragma unroll
        for (int a = 0; a < KNN; ++a)
#pragma unroll
            for (int c2 = a + 1; c2 < KNN; ++c2) {
                float dx = p[a].x - p[c2].x, dy = p[a].y - p[c2].y, dz = p[a].z - p[c2].z;
                sum += sqrtf(dx * dx + dy * dy + dz * dz);
            }
        const int q = qbase + lm;
        dens[(size_t)b * NPT + q] = 2.0f * sum;
#pragma unroll
        for (int s = 0; s < KNN; ++s) nbr[((size_t)b * NPT + q) * KNN + s] = sel[s];
    }
}

// ---------------- stable rank (== argsort ascending) + output ----------------

// Grid: (NPT/256, NB), block 256.
__global__ void gp_rank_out(const float4* __restrict__ cand4,
                            const float* __restrict__ dens,
                            const int* __restrict__ nbr,
                            const float* __restrict__ sig,
                            float* __restrict__ outc,
                            float* __restrict__ outs) {
    __shared__ float sd[2048];
    const int b = blockIdx.y;
    const int i = blockIdx.x * 256 + threadIdx.x;
    const float* db = dens + (size_t)b * NPT;
    const float di = db[i];
    int rank = 0;
    for (int cb = 0; cb < NPT; cb += 2048) {
        __syncthreads();
        for (int t = threadIdx.x; t < 2048; t += 256) sd[t] = db[cb + t];
        __syncthreads();
        const float4* sd4 = (const float4*)sd;
        for (int j4 = 0; j4 < 512; ++j4) {
            float4 v = sd4[j4];
            int jj = cb + j4 * 4;
            rank += ((v.x < di) || (v.x == di && (jj + 0) < i)) ? 1 : 0;
            rank += ((v.y < di) || (v.y == di && (jj + 1) < i)) ? 1 : 0;
            rank += ((v.z < di) || (v.z == di && (jj + 2) < i)) ? 1 : 0;
            rank += ((v.w < di) || (v.w == di && (jj + 3) < i)) ? 1 : 0;
        }
    }
    if (rank < NKEEP) {
        float4 p = cand4[(size_t)b * NPT + i];
        float* oc = outc + ((size_t)b * NKEEP + rank) * 3;
        oc[0] = p.x; oc[1] = p.y; oc[2] = p.z;

        const int* nb = nbr + ((size_t)b * NPT + i) * KNN;
        const float* sb = sig + (size_t)b * NPT * 64;
        const float4* s0 = (const float4*)(sb + (size_t)nb[0] * 64);
        const float4* s1 = (const float4*)(sb + (size_t)nb[1] * 64);
        const float4* s2 = (const float4*)(sb + (size_t)nb[2] * 64);
        const float4* s3 = (const float4*)(sb + (size_t)nb[3] * 64);
        const float4* s4 = (const float4*)(sb + (size_t)nb[4] * 64);
        float4* os = (float4*)(outs + ((size_t)b * NKEEP + rank) * 64);
#pragma unroll
        for (int d = 0; d < 16; ++d) {
            float4 a = s0[d], c = s1[d], e = s2[d], f = s3[d], g = s4[d];
            float4 o;
            o.x = (a.x + c.x + e.x + f.x + g.x) * 0.2f;
            o.y = (a.y + c.y + e.y + f.y + g.y) * 0.2f;
            o.z = (a.z + c.z + e.z + f.z + g.z) * 0.2f;
            o.w = (a.w + c.w + e.w + f.w + g.w) * 0.2f;
            os[d] = o;
        }
    }
}

// ---------------- launch ----------------

extern "C" void kernel_launch(void* const* d_in, const int* in_sizes, int n_in,
                              void* d_out, int out_size, void* d_ws, size_t ws_size,
                              hipStream_t stream) {
    (void)in_sizes; (void)n_in; (void)out_size; (void)ws_size;
    const float* coords = (const float*)d_in[0];   // [4,8192,3]
    const float* signal = (const float*)d_in[1];   // [4,8192,64]

    char* ws = (char*)d_ws;
    float4* cand4 = (float4*)(ws);                         // 4*8192*16  = 524288 B
    float*  cent  = (float*)(ws + 524288);                 // 4*4*4      = 64 B
    float*  dens  = (float*)(ws + 524352);                 // 4*8192*4   = 131072 B
    int*    nbr   = (int*)(ws + 655424);                   // 4*8192*5*4 = 655360 B

    float* outc = (float*)d_out;                           // [4,4096,3]
    float* outs = outc + (size_t)NB * NKEEP * 3;           // [4,4096,64]

    dim3 upd(NPT / 256, NB);

    // gravity iteration 1 (reads input, writes packed {x,y,z,|c|^2})
    gp_centroid_in<<<NB, 256, 0, stream>>>(coords, cent);
    gp_update_in  <<<upd, 256, 0, stream>>>(coords, cent, cand4);
    // gravity iterations 2,3 (in place)
    gp_centroid4<<<NB, 256, 0, stream>>>(cand4, cent);
    gp_update4  <<<upd, 256, 0, stream>>>(cand4, cent);
    gp_centroid4<<<NB, 256, 0, stream>>>(cand4, cent);
    gp_update4  <<<upd, 256, 0, stream>>>(cand4, cent);

    // WMMA kNN + density (async double-buffered candidate streaming)
    gp_topk<<<dim3(NPT / 128, NB), 256, 0, stream>>>(cand4, dens, nbr);

    // stable rank + gather/pool outputs
    gp_rank_out<<<dim3(NPT / 256, NB), 256, 0, stream>>>(cand4, dens, nbr, signal, outc, outs);
}